// GCN1PoolNorm_74543452389560
// MI455X (gfx1250) — compile-verified
//
#include <hip/hip_runtime.h>
#include <hip/hip_bf16.h>
#include <math.h>

typedef __attribute__((ext_vector_type(2))) float v2f;
typedef __attribute__((ext_vector_type(8))) float v8f;

#define HID 64
#define NGRAPH 64

__device__ __forceinline__ void atomAddF(float* p, float v) {
    unsafeAtomicAdd(p, v);   // lowers to global_atomic_add_f32 on gfx1250
}

// ---------------------------------------------------------------- degree
__global__ void degree_k(const int* __restrict__ dst, float* __restrict__ deg, int nE) {
    int i = blockIdx.x * blockDim.x + threadIdx.x;
    if (i < nE) atomAddF(&deg[dst[i]], 1.0f);
}

// dis = rsqrt(deg+1) (in place), self_coef = dis*dis
__global__ void node_norm_k(float* __restrict__ deg_dis, float* __restrict__ selfc, int nN) {
    int i = blockIdx.x * blockDim.x + threadIdx.x;
    if (i < nN) {
        float d = rsqrtf(deg_dis[i] + 1.0f);
        deg_dis[i] = d;
        selfc[i] = d * d;
    }
}

__global__ void edge_coef_k(const int* __restrict__ src, const int* __restrict__ dst,
                            const float* __restrict__ dis, float* __restrict__ coef, int nE) {
    int i = blockIdx.x * blockDim.x + threadIdx.x;
    if (i < nE) coef[i] = dis[src[i]] * dis[dst[i]];
}

// ---------------------------------------------------------------- WMMA GEMM: H = A[nN x 64] * W[64 x 64]
// One wave per 16-row tile; 4 accumulators cover all 64 output columns.
__global__ void gemm64_wmma(const float* __restrict__ A, const float* __restrict__ Wt,
                            float* __restrict__ H, int mTiles) {
    int wave = threadIdx.x >> 5;
    int lane = threadIdx.x & 31;
    int mt = blockIdx.x * 4 + wave;
    if (mt >= mTiles) return;                 // uniform per-wave: EXEC stays all-ones
    int m0   = mt * 16;
    int half = lane >> 4;                     // 0: K=k0..k0+1, 1: K=k0+2..k0+3
    int lmod = lane & 15;                     // A: row;  B/C/D: column
    const float* Arow = A + (size_t)(m0 + lmod) * HID;
    v8f c0 = {}, c1 = {}, c2 = {}, c3 = {};
#pragma unroll
    for (int k0 = 0; k0 < HID; k0 += 4) {
        int k = k0 + 2 * half;
        v2f a;  a.x = Arow[k];  a.y = Arow[k + 1];
        const float* Wk = Wt + (size_t)k * HID + lmod;
        v2f b0, b1, b2, b3;
        b0.x = Wk[0];  b0.y = Wk[HID];        // rows k, k+1 ; cols lmod+{0,16,32,48}
        b1.x = Wk[16]; b1.y = Wk[HID + 16];
        b2.x = Wk[32]; b2.y = Wk[HID + 32];
        b3.x = Wk[48]; b3.y = Wk[HID + 48];
        c0 = __builtin_amdgcn_wmma_f32_16x16x4_f32(false, a, false, b0, (short)0, c0, false, false);
        c1 = __builtin_amdgcn_wmma_f32_16x16x4_f32(false, a, false, b1, (short)0, c1, false, false);
        c2 = __builtin_amdgcn_wmma_f32_16x16x4_f32(false, a, false, b2, (short)0, c2, false, false);
        c3 = __builtin_amdgcn_wmma_f32_16x16x4_f32(false, a, false, b3, (short)0, c3, false, false);
    }
#pragma unroll
    for (int r = 0; r < 8; ++r) {
        size_t row = (size_t)(m0 + r + 8 * half) * HID + lmod;
        H[row]      = c0[r];
        H[row + 16] = c1[r];
        H[row + 32] = c2[r];
        H[row + 48] = c3[r];
    }
}

// ---------------------------------------------------------------- edge gather/scatter (L2-resident)
// One wave per edge, each lane handles 2 features (float2).
__global__ void edge_scatter(const float* __restrict__ h, const int* __restrict__ src,
                             const int* __restrict__ dst, const float* __restrict__ coef,
                             float* __restrict__ agg, int nE) {
    int e = blockIdx.x * 8 + (threadIdx.x >> 5);
    if (e >= nE) return;
    int lane = threadIdx.x & 31;
    int s = src[e], d = dst[e];
    float c = coef[e];
    int f = lane * 2;
    const float2 v = *(const float2*)(h + (size_t)s * HID + f);
    float* out = agg + (size_t)d * HID + f;
    atomAddF(out,     v.x * c);
    atomAddF(out + 1, v.y * c);
}

// y = agg + h*self_coef + bias (in place on agg) and accumulate per-feature sum / sumsq
__global__ void self_bias_stats(float* __restrict__ y, const float* __restrict__ h,
                                const float* __restrict__ selfc, const float* __restrict__ bias,
                                float* __restrict__ stats, int nN) {
    __shared__ float s_sum[256], s_sq[256];
    int f  = threadIdx.x & 63;
    int rl = threadIdx.x >> 6;                       // 0..3
    float bf = bias[f];
    float sum = 0.f, sq = 0.f;
    for (int row = blockIdx.x * 4 + rl; row < nN; row += gridDim.x * 4) {
        size_t idx = (size_t)row * HID + f;
        float v = y[idx] + h[idx] * selfc[row] + bf;
        y[idx] = v;
        sum += v; sq += v * v;
    }
    s_sum[threadIdx.x] = sum; s_sq[threadIdx.x] = sq;
    __syncthreads();
    if (threadIdx.x < 64) {
        float ts = s_sum[threadIdx.x] + s_sum[threadIdx.x + 64] + s_sum[threadIdx.x + 128] + s_sum[threadIdx.x + 192];
        float tq = s_sq[threadIdx.x]  + s_sq[threadIdx.x + 64]  + s_sq[threadIdx.x + 128]  + s_sq[threadIdx.x + 192];
        atomAddF(&stats[threadIdx.x], ts);
        atomAddF(&stats[64 + threadIdx.x], tq);
    }
}

__global__ void bn_finalize(const float* __restrict__ stats, const float* __restrict__ gamma,
                            const float* __restrict__ beta, float* __restrict__ scsh, float invN) {
    int f = threadIdx.x;                              // 64 threads
    float mu  = stats[f] * invN;
    float var = stats[64 + f] * invN - mu * mu;
    float sc  = gamma[f] * rsqrtf(var + 1e-5f);
    scsh[f]      = sc;
    scsh[64 + f] = beta[f] - mu * sc;
}

__global__ void bn_relu(const float* __restrict__ y, const float* __restrict__ scsh,
                        float* __restrict__ out, int total) {
    int i = blockIdx.x * blockDim.x + threadIdx.x;
    if (i < total) {
        int f = i & 63;
        float v = fmaf(y[i], scsh[f], scsh[64 + f]);
        out[i] = v > 0.f ? v : 0.f;
    }
}

// ---------------------------------------------------------------- global max pool (sorted equal segments)
__global__ void pool_max(const float* __restrict__ x, float* __restrict__ emb, int nN) {
    int g = blockIdx.x;
    int start = (g * nN + NGRAPH - 1) / NGRAPH;       // first i with (i*NG)//nN == g
    int end   = ((g + 1) * nN + NGRAPH - 1) / NGRAPH;
    int f  = threadIdx.x & 63;
    int rl = threadIdx.x >> 6;
    float m = -INFINITY;
    for (int row = start + rl; row < end; row += 4)
        m = fmaxf(m, x[(size_t)row * HID + f]);
    __shared__ float s[256];
    s[threadIdx.x] = m;
    __syncthreads();
    if (threadIdx.x < 64) {
        float r = fmaxf(fmaxf(s[threadIdx.x], s[threadIdx.x + 64]),
                        fmaxf(s[threadIdx.x + 128], s[threadIdx.x + 192]));
        emb[g * HID + threadIdx.x] = r;
    }
}

// ---------------------------------------------------------------- tiny MLP head: relu(emb@W1+b1)@W2+b2
__global__ void mlp_head(const float* __restrict__ emb, const float* __restrict__ w1,
                         const float* __restrict__ b1, const float* __restrict__ w2,
                         const float* __restrict__ b2, float* __restrict__ out) {
    __shared__ float hid[NGRAPH * HID];
    for (int o = threadIdx.x; o < NGRAPH * HID; o += blockDim.x) {
        int g = o >> 6, f = o & 63;
        float acc = b1[f];
        for (int k = 0; k < HID; ++k) acc = fmaf(emb[g * HID + k], w1[k * HID + f], acc);
        hid[o] = acc > 0.f ? acc : 0.f;
    }
    __syncthreads();
    for (int o = threadIdx.x; o < NGRAPH * 10; o += blockDim.x) {
        int g = o / 10, c = o % 10;
        float acc = b2[c];
        for (int k = 0; k < HID; ++k) acc = fmaf(hid[g * HID + k], w2[k * 10 + c], acc);
        out[o] = acc;
    }
}

// ----------------------------------------------------------------
extern "C" void kernel_launch(void* const* d_in, const int* in_sizes, int n_in,
                              void* d_out, int out_size, void* d_ws, size_t ws_size,
                              hipStream_t stream) {
    const float* x    = (const float*)d_in[0];
    const int*   ei   = (const int*)d_in[1];
    const float* Wl[3] = {(const float*)d_in[3], (const float*)d_in[5], (const float*)d_in[7]};
    const float* bl[3] = {(const float*)d_in[4], (const float*)d_in[6], (const float*)d_in[8]};
    const float* gamma = (const float*)d_in[9];
    const float* beta  = (const float*)d_in[10];
    const float* l1w = (const float*)d_in[11];
    const float* l1b = (const float*)d_in[12];
    const float* l2w = (const float*)d_in[13];
    const float* l2b = (const float*)d_in[14];

    const int nN = in_sizes[0] / HID;         // 50000
    const int nE = in_sizes[1] / 2;           // 800000
    const int* src = ei;
    const int* dst = ei + nE;

    // workspace layout (floats)
    float* wsf   = (float*)d_ws;
    float* dis   = wsf;                       // nN (deg -> dis in place)
    float* selfc = dis   + nN;                // nN
    float* coef  = selfc + nN;                // nE
    float* h     = coef  + nE;                // nN*64
    float* agg   = h     + (size_t)nN * HID;  // nN*64
    float* bufA  = agg   + (size_t)nN * HID;  // nN*64
    float* bufB  = bufA  + (size_t)nN * HID;  // nN*64
    float* stats = bufB  + (size_t)nN * HID;  // 128
    float* scsh  = stats + 128;               // 128
    float* emb   = scsh  + 128;               // 64*64

    // degree + normalization coefficients
    hipMemsetAsync(dis, 0, (size_t)nN * sizeof(float), stream);
    degree_k   <<<(nE + 255) / 256, 256, 0, stream>>>(dst, dis, nE);
    node_norm_k<<<(nN + 255) / 256, 256, 0, stream>>>(dis, selfc, nN);
    edge_coef_k<<<(nE + 255) / 256, 256, 0, stream>>>(src, dst, dis, coef, nE);

    const float* cur = x;
    float* pingpong[2] = {bufA, bufB};
    const int mTiles = nN / 16;               // 50000 % 16 == 0
    for (int l = 0; l < 3; ++l) {
        hipMemsetAsync(agg,   0, (size_t)nN * HID * sizeof(float), stream);
        hipMemsetAsync(stats, 0, 128 * sizeof(float), stream);
        gemm64_wmma    <<<(mTiles + 3) / 4, 128, 0, stream>>>(cur, Wl[l], h, mTiles);
        edge_scatter   <<<(nE + 7) / 8,     256, 0, stream>>>(h, src, dst, coef, agg, nE);
        self_bias_stats<<<512,              256, 0, stream>>>(agg, h, selfc, bl[l], stats, nN);
        bn_finalize    <<<1,                 64, 0, stream>>>(stats, gamma, beta, scsh, 1.0f / (float)nN);
        float* nxt = pingpong[l & 1];
        bn_relu        <<<((size_t)nN * HID + 255) / 256, 256, 0, stream>>>(agg, scsh, nxt, nN * HID);
        cur = nxt;
    }

    pool_max<<<NGRAPH, 256, 0, stream>>>(cur, emb, nN);
    mlp_head<<<1,      256, 0, stream>>>(emb, l1w, l1b, l2w, l2b, (float*)d_out);
}